// NaryDisEmbedding_71279277245039
// MI455X (gfx1250) — compile-verified
//
#include <hip/hip_runtime.h>

typedef __attribute__((ext_vector_type(2))) float v2f;
typedef __attribute__((ext_vector_type(8))) float v8f;

#define EMBED_DIM 128
#define OUT_DIM   256

// out[row, 0:256] = [c2_0,c2_1]@emb2 ++ [c3_0,c3_1,c3_2]@emb3
// folded into a K=4 f32 WMMA:  A=[c2_1, c3_1, c3_2, 1],  B = transformed table.
__global__ __launch_bounds__(256) void NaryDisEmbedding_wmma_kernel(
    const int* __restrict__ x,
    const float* __restrict__ emb2,
    const float* __restrict__ emb3,
    float* __restrict__ out)
{
    __shared__ float T[4][OUT_DIM];

    const int tid = threadIdx.x;

    // Build the transformed 4x256 B-table once per block (one column per thread).
    {
        const int d = tid;                       // 0..255
        float t0, t1, t2, t3;
        if (d < EMBED_DIM) {
            const float e0 = emb2[d];
            const float e1 = emb2[EMBED_DIM + d];
            t0 = e1 - e0; t1 = 0.0f; t2 = 0.0f; t3 = 16.0f * e0;
        } else {
            const int dd = d - EMBED_DIM;
            const float e0 = emb3[dd];
            const float e1 = emb3[EMBED_DIM + dd];
            const float e2 = emb3[2 * EMBED_DIM + dd];
            t0 = 0.0f; t1 = e1 - e0; t2 = e2 - e0; t3 = 16.0f * e0;
        }
        T[0][d] = t0; T[1][d] = t1; T[2][d] = t2; T[3][d] = t3;
    }
    __syncthreads();

    const int wave  = tid >> 5;       // 8 waves / block
    const int lane  = tid & 31;
    const int col16 = lane & 15;
    const bool hi   = lane >= 16;     // lane half selects K0/K1 vs K2/K3 slots

    const int rowBase = (blockIdx.x * 8 + wave) * 16;

    // Each lane computes digit counts for row rowBase + (lane&15).
    // x stream is read exactly once -> non-temporal load.
    const int xv = __builtin_nontemporal_load(&x[rowBase + col16]);
    const int c2 = __popc(xv);        // x < 2^16, count of binary 1-digits
    int c31 = 0, c32 = 0;
    int t = xv;                        // x < 3^10 => 10 ternary digits suffice
#pragma unroll
    for (int i = 0; i < 10; ++i) {
        const int q  = t / 3;          // compiler -> magic multiply
        const int dg = t - q * 3;
        t = q;
        c31 += (dg == 1);
        c32 += (dg == 2);
    }

    // A-matrix 16x4 f32 layout: VGPR0 = K0 (lanes 0-15) / K2 (lanes 16-31),
    //                           VGPR1 = K1 / K3.
    v2f a;
    a.x = hi ? (float)c32 : (float)c2;   // K2 : K0
    a.y = hi ? 1.0f       : (float)c31;  // K3 : K1

    float* outRow = out + (size_t)rowBase * OUT_DIM;

#pragma unroll
    for (int tile = 0; tile < 16; ++tile) {
        const int col = tile * 16 + col16;
        // B-matrix 4x16 f32: VGPR0 = K0/K2 across lane halves, VGPR1 = K1/K3.
        v2f b;
        b.x = T[hi ? 2 : 0][col];
        b.y = T[hi ? 3 : 1][col];
        v8f c = {};
        // 8 args: (neg_a, A, neg_b, B, c_mod, C, reuse_a, reuse_b)
        v8f d = __builtin_amdgcn_wmma_f32_16x16x4_f32(
            false, a, false, b, (short)0, c, false, false);
        // D layout: VGPR j holds row j (lanes 0-15) / row j+8 (lanes 16-31),
        // N = lane % 16. Output is an 800MB write-once stream (>> 192MB L2):
        // non-temporal stores avoid L2 thrash.
#pragma unroll
        for (int j = 0; j < 8; ++j) {
            const int r = j + (hi ? 8 : 0);
            __builtin_nontemporal_store(d[j], &outRow[r * OUT_DIM + col]);
        }
    }
}

extern "C" void kernel_launch(void* const* d_in, const int* in_sizes, int n_in,
                              void* d_out, int out_size, void* d_ws, size_t ws_size,
                              hipStream_t stream) {
    const int*   x    = (const int*)d_in[0];    // [4096*200] int32
    const float* emb2 = (const float*)d_in[1];  // [2,128] f32
    const float* emb3 = (const float*)d_in[2];  // [3,128] f32
    float*       out  = (float*)d_out;          // [4096*200, 256] f32

    const int n_rows = in_sizes[0];             // 819200, divisible by 128
    const int blocks = n_rows / 128;            // 8 waves x 16 rows per block
    NaryDisEmbedding_wmma_kernel<<<blocks, 256, 0, stream>>>(x, emb2, emb3, out);
}